// SSFM_61727269978266
// MI455X (gfx1250) — compile-verified
//
#include <hip/hip_runtime.h>
#include <cstdint>
#include <cmath>

// ---------------------------------------------------------------------------
// SSFM on MI455X (gfx1250).
//  - N = 2^18 split as 512 x 512 four-step FFT, digit-permuted (no transposes).
//  - Each 512-pt FFT: radix-2 DIF + two 256-pt FFTs done as W16*U*W16 complex
//    matmuls on V_WMMA_F32_16X16X4_F32 (full f32 precision, complex64 data).
//  - Twiddles come from two 512-entry LDS tables (exact 2pi/2^k angles),
//    built once per WG and overlapped with the TDM tile DMA; only dispersion
//    and Kerr phases use sincosf.
//  - Column tiles staged via the Tensor Data Mover (tensor_load_to_lds) with
//    s_wait_tensorcnt; row tiles via 1D TDM load.
//  - 131 launches: KCfirst + 65*KR + 64*KCmid + KClast; everything L2-resident.
// ---------------------------------------------------------------------------

#define NTOT   262144
#define BN     1048576   // BATCH*NTOT

typedef float v2f  __attribute__((ext_vector_type(2)));
typedef float v8f  __attribute__((ext_vector_type(8)));
typedef unsigned int u32x4 __attribute__((ext_vector_type(4)));
typedef int   i32x4 __attribute__((ext_vector_type(4)));
typedef int   i32x8 __attribute__((ext_vector_type(8)));

__device__ __forceinline__ v8f wmma_f32(v2f a, v2f b, v8f c) {
#if defined(__AMDGCN__)
  // (neg_a, A, neg_b, B, c_mod, C, reuse_a, reuse_b); A/B neg must be 0 for F32
  return __builtin_amdgcn_wmma_f32_16x16x4_f32(false, a, false, b, (short)0, c,
                                               false, false);
#else
  return c;
#endif
}

// TDM: build D# groups per CDNA5 ISA ch.8 and issue tensor_load_to_lds.
// Elements are 8-byte (complex64) -> data_size = 3.  6-arg builtin form.
__device__ __forceinline__ void tdm_load(uint32_t lds_off, const void* g,
                                         uint32_t tile_w, uint32_t tile_h,
                                         uint32_t stride0, uint32_t tens_w,
                                         uint32_t tens_h) {
#if defined(__AMDGCN__)
  uint64_t ga = (uint64_t)(uintptr_t)g;
  u32x4 g0;
  g0[0] = 1u;                                    // count=1, user descriptor
  g0[1] = lds_off;                               // lds_addr [63:32]
  g0[2] = (uint32_t)ga;                          // global_addr lo
  g0[3] = ((uint32_t)(ga >> 32) & 0x01FFFFFFu) | (2u << 30);  // addr hi | type=2
  i32x8 g1;
  g1[0] = (int)(3u << 16);                       // data_size=3 (8B), mask=0
  g1[1] = (int)((tens_w & 0xFFFFu) << 16);       // tensor_dim0[15:0]
  g1[2] = (int)(((tens_w >> 16) & 0xFFFFu) | ((tens_h & 0xFFFFu) << 16));
  g1[3] = (int)(((tens_h >> 16) & 0xFFFFu) | ((tile_w & 0xFFFFu) << 16));
  g1[4] = (int)(tile_h & 0xFFFFu);               // tile_dim1 (tile_dim2=0)
  g1[5] = (int)stride0;                          // tensor_dim0_stride lo
  g1[6] = 0;
  g1[7] = 0;
  i32x4 z4; z4[0] = z4[1] = z4[2] = z4[3] = 0;
  i32x8 z8; z8[0] = z8[1] = z8[2] = z8[3] = z8[4] = z8[5] = z8[6] = z8[7] = 0;
  __builtin_amdgcn_tensor_load_to_lds(g0, g1, z4, z4, z8, 0);
#endif
}

__device__ __forceinline__ void wait_tensor0() {
#if defined(__AMDGCN__)
  __builtin_amdgcn_s_wait_tensorcnt(0);
#endif
}

__device__ __forceinline__ void wbar() {
#if defined(__AMDGCN__)
  __builtin_amdgcn_wave_barrier();
#endif
}

// Fill twiddle tables: t512[j] = e^{+i*2pi*j/512}, tN[j] = e^{+i*2pi*j/2^18}.
__device__ __forceinline__ void fill_tables(float2* t512, float2* tN, int t) {
  const float TP = 6.28318530717958648f;
  for (int i = t; i < 512; i += 256) {
    float s, c;
    sincosf(TP * (float)i * (1.0f / 512.0f), &s, &c);
    t512[i] = make_float2(c, s);
    sincosf(TP * (float)i * (1.0f / 262144.0f), &s, &c);
    tN[i] = make_float2(c, s);
  }
}

// q * (w.x + i*sgn*w.y)
__device__ __forceinline__ float2 cmul_tw(float2 q, float2 w, float sgn) {
  float s = sgn * w.y;
  return make_float2(q.x * w.x - q.y * s, q.x * s + q.y * w.x);
}

// e^{+i*2pi*a/2^18} via a = 512*q + r  (a in [0, 2^18))
__device__ __forceinline__ float2 twiddleN(const float2* t512, const float2* tN,
                                           int a) {
  float2 A = t512[(a >> 9) & 511];
  float2 B = tN[a & 511];
  return make_float2(A.x * B.x - A.y * B.y, A.x * B.y + A.y * B.x);
}

__device__ __forceinline__ float winval(int n) {
  // periodic Hamming half-windows at both ends, length 100 each
  if (n < 100)         return 0.54f - 0.46f * cosf(0.0314159265358979f * (float)n);
  if (n >= NTOT - 100) return 0.54f - 0.46f * cosf(0.0314159265358979f * (float)(n - NTOT + 200));
  return 1.0f;
}

// ---------------------------------------------------------------------------
// 512-point DFT (one wave32), natural order in/out, unnormalized.
// sgn = -1: forward (e^{-i}), sgn = +1: inverse kernel (e^{+i}).
// Element j lives at p[j*stride] in LDS.  Structure:
//   radix-2 DIF -> two 256-pt FFTs, each = (W16 * U) twiddle (* W16) via WMMA.
// All twiddles are exact multiples of 2pi/512 -> read from t512.
// ---------------------------------------------------------------------------
__device__ __forceinline__ void wave_fft512(float2* p, int stride, float sgn,
                                            int lane, const float2* t512) {
  // ---- radix-2 DIF stage (u -> [0,256), v*w512^j -> [256,512)) -------------
  for (int t = lane; t < 256; t += 32) {
    float2 a = p[t * stride];
    float2 b = p[(t + 256) * stride];
    float dr = a.x - b.x, di = a.y - b.y;
    float2 w = t512[t];
    float c = w.x, s = sgn * w.y;
    p[t * stride]         = make_float2(a.x + b.x, a.y + b.y);
    p[(t + 256) * stride] = make_float2(dr * c - di * s, dr * s + di * c);
  }
  wbar();

  // ---- W16 twiddle chunks in both A and B lane layouts (identical, W sym) --
  const int n16   = lane & 15;
  const int khalf = (lane >> 4) << 1;          // 0 or 2
  const int mofs  = (lane < 16) ? 0 : 8;       // C/D row offset
  v2f wc[4], wsp[4], wsn[4];                    // cos, +sgn*sin, -sgn*sin
#pragma unroll
  for (int j = 0; j < 4; ++j) {
    int k0 = 4 * j + khalf;
    float2 w0 = t512[((n16 * k0) & 15) << 5];        // angle*2pi/16
    float2 w1 = t512[((n16 * (k0 + 1)) & 15) << 5];
    wc[j][0]  = w0.x;        wc[j][1]  = w1.x;
    wsp[j][0] = sgn * w0.y;  wsp[j][1] = sgn * w1.y;
    wsn[j][0] = -sgn * w0.y; wsn[j][1] = -sgn * w1.y;
  }

  v8f yr[2], yi[2];
  // ---- stage A: T = W16 * U, mid twiddle, store T' row-major back ----------
#pragma unroll
  for (int h = 0; h < 2; ++h) {
    const int base = h << 8;
    v8f ar = {0, 0, 0, 0, 0, 0, 0, 0};
    v8f ai = {0, 0, 0, 0, 0, 0, 0, 0};
#pragma unroll
    for (int j = 0; j < 4; ++j) {
      int kk = 4 * j + khalf;
      float2 u0 = p[(base + kk * 16 + n16) * stride];
      float2 u1 = p[(base + (kk + 1) * 16 + n16) * stride];
      v2f br, bi;
      br[0] = u0.x; br[1] = u1.x;
      bi[0] = u0.y; bi[1] = u1.y;
      ar = wmma_f32(wc[j],  br, ar);    // + cos * Ur
      ar = wmma_f32(wsn[j], bi, ar);    // - (sgn sin) * Ui
      ai = wmma_f32(wc[j],  bi, ai);    // + cos * Ui
      ai = wmma_f32(wsp[j], br, ai);    // + (sgn sin) * Ur
    }
#pragma unroll
    for (int r = 0; r < 8; ++r) {       // C layout: row = r+mofs, col = n16
      int mm = r + mofs;
      float2 w = t512[((mm * n16) & 255) << 1];      // angle*2pi/256
      float c = w.x, s = sgn * w.y;
      float tr = ar[r] * c - ai[r] * s;
      float ti = ar[r] * s + ai[r] * c;
      p[(base + mm * 16 + n16) * stride] = make_float2(tr, ti);
    }
  }
  wbar();
  // ---- stage B: Y = T' * W16 ----------------------------------------------
#pragma unroll
  for (int h = 0; h < 2; ++h) {
    const int base = h << 8;
    v8f cr = {0, 0, 0, 0, 0, 0, 0, 0};
    v8f ci = {0, 0, 0, 0, 0, 0, 0, 0};
#pragma unroll
    for (int j = 0; j < 4; ++j) {
      int k0 = 4 * j + khalf;
      float2 t0 = p[(base + n16 * 16 + k0) * stride];
      float2 t1 = p[(base + n16 * 16 + k0 + 1) * stride];
      v2f aR, aI;
      aR[0] = t0.x; aR[1] = t1.x;
      aI[0] = t0.y; aI[1] = t1.y;
      cr = wmma_f32(aR, wc[j],  cr);
      cr = wmma_f32(aI, wsn[j], cr);
      ci = wmma_f32(aR, wsp[j], ci);
      ci = wmma_f32(aI, wc[j],  ci);
    }
    yr[h] = cr; yi[h] = ci;
  }
  wbar();
  // ---- scatter to natural order: X[2*(k1+16*k2)+h] ------------------------
#pragma unroll
  for (int h = 0; h < 2; ++h)
#pragma unroll
    for (int r = 0; r < 8; ++r) {
      int k1  = r + mofs;
      int pos = ((k1 + 16 * n16) << 1) + h;
      p[pos * stride] = make_float2(yr[h][r], yi[h][r]);
    }
  wbar();
}

// ---------------------------------------------------------------------------
// Kernels. Grid = 256 WGs of 256 threads (8 wave32 per WG); each wave owns one
// 512-pt FFT. LDS: 32 KB tile + 8 KB twiddle tables.
// ---------------------------------------------------------------------------

// First pass: load input, forward column FFT over n1, outer fwd twiddle, store.
__global__ __launch_bounds__(256) void kc_first(const float* __restrict__ xr,
                                                const float* __restrict__ xi,
                                                float2* __restrict__ ws) {
  __shared__ float2 tile[4096];
  __shared__ float2 t512[512];
  __shared__ float2 tN[512];
  const int t = threadIdx.x, lane = t & 31, wave = t >> 5;
  const int blk = blockIdx.x, b = blk >> 6, col0 = (blk & 63) << 3;
  const size_t gb = (size_t)b << 18;
  fill_tables(t512, tN, t);
  for (int i = t; i < 4096; i += 256) {
    int row = i >> 3, c = i & 7;
    int n = (row << 9) + col0 + c;
    tile[i] = make_float2(xr[gb + n], xi[gb + n]);
  }
  __syncthreads();
  float2* p = tile + wave;
  const int col = col0 + wave;
  wave_fft512(p, 8, -1.0f, lane, t512);
  for (int j = lane; j < 512; j += 32)   // * w_N^{-n2*k1}
    p[j * 8] = cmul_tw(p[j * 8], twiddleN(t512, tN, (j * col) & (NTOT - 1)), -1.0f);
  __syncthreads();
  for (int i = t; i < 4096; i += 256) {
    int row = i >> 3, c = i & 7;
    ws[gb + ((size_t)row << 9) + col0 + c] = tile[i];
  }
}

// Row pass: fwd 512-FFT over n2, dispersion at k=k1+512*k2, inv 512-FFT,
// inverse outer twiddle. Rows are contiguous: 1D TDM load of 32 KB.
__global__ __launch_bounds__(256) void kr_step(float2* __restrict__ ws,
                                               float dcoef) {
  __shared__ float2 tile[4096];
  __shared__ float2 t512[512];
  __shared__ float2 tN[512];
  const int t = threadIdx.x, lane = t & 31, wave = t >> 5;
  const int blk = blockIdx.x, b = blk >> 6, r0 = (blk & 63) << 3;
  float2* gbase = ws + ((size_t)b << 18) + ((size_t)r0 << 9);
  if (t < 32)
    tdm_load((uint32_t)(uintptr_t)&tile[0], gbase, 4096u, 1u, 4096u, 4096u, 1u);
  fill_tables(t512, tN, t);                      // overlap with DMA
  if (t < 32) wait_tensor0();
  __syncthreads();
  float2* p = tile + (wave << 9);
  const int k1 = r0 + wave;
  wave_fft512(p, 1, -1.0f, lane, t512);
  for (int k2 = lane; k2 < 512; k2 += 32) {
    int kk = k1 + (k2 << 9);
    int ks = (kk >= 131072) ? kk - 262144 : kk;
    float fr = (float)ks * (1.0f / 262144.0f);
    float ph = dcoef * fr * fr;                  // dispersion e^{i*phase}
    float s, c; sincosf(ph, &s, &c);
    float2 q = p[k2];
    p[k2] = make_float2(q.x * c - q.y * s, q.x * s + q.y * c);
  }
  wave_fft512(p, 1, 1.0f, lane, t512);
  for (int j = lane; j < 512; j += 32)           // * w_N^{+n2*k1}
    p[j] = cmul_tw(p[j], twiddleN(t512, tN, (j * k1) & (NTOT - 1)), 1.0f);
  __syncthreads();
  float4* g4 = (float4*)gbase;
  float4* t4 = (float4*)tile;
  for (int i = t; i < 2048; i += 256) g4[i] = t4[i];
}

// Column pass: inverse 512-FFT over k1 (ends previous linear op), then
// scale(=atten/N), window, Kerr phase, then forward 512-FFT + fwd twiddle
// (starts next linear op). 2D strided TDM tile load.
__global__ __launch_bounds__(256) void kc_mid(float2* __restrict__ ws,
                                              float scale, float nlc) {
  __shared__ float2 tile[4096];
  __shared__ float2 t512[512];
  __shared__ float2 tN[512];
  const int t = threadIdx.x, lane = t & 31, wave = t >> 5;
  const int blk = blockIdx.x, b = blk >> 6, col0 = (blk & 63) << 3;
  const size_t gb = (size_t)b << 18;
  if (t < 32)
    tdm_load((uint32_t)(uintptr_t)&tile[0], ws + gb + col0, 8u, 512u, 512u, 512u, 512u);
  fill_tables(t512, tN, t);                      // overlap with DMA
  if (t < 32) wait_tensor0();
  __syncthreads();
  float2* p = tile + wave;
  const int col = col0 + wave;
  wave_fft512(p, 8, 1.0f, lane, t512);           // inverse over k1 -> time
  for (int j = lane; j < 512; j += 32) {
    int n = (j << 9) + col;
    float2 q = p[j * 8];
    float w = winval(n) * scale;
    float re = q.x * w, im = q.y * w;
    float ph = nlc * (re * re + im * im);        // Kerr: e^{-i*gamma*dz*|q|^2}
    float s, c; sincosf(ph, &s, &c);
    p[j * 8] = make_float2(re * c - im * s, re * s + im * c);
  }
  wave_fft512(p, 8, -1.0f, lane, t512);          // forward over n1
  for (int j = lane; j < 512; j += 32)
    p[j * 8] = cmul_tw(p[j * 8], twiddleN(t512, tN, (j * col) & (NTOT - 1)), -1.0f);
  __syncthreads();
  for (int i = t; i < 4096; i += 256) {
    int row = i >> 3, c = i & 7;
    ws[gb + ((size_t)row << 9) + col0 + c] = tile[i];
  }
}

// Final pass: inverse column FFT, apply atten/N, write planar [re | im].
__global__ __launch_bounds__(256) void kc_last(float2* __restrict__ ws,
                                               float* __restrict__ out,
                                               float scale) {
  __shared__ float2 tile[4096];
  __shared__ float2 t512[512];
  __shared__ float2 tN[512];
  const int t = threadIdx.x, lane = t & 31, wave = t >> 5;
  const int blk = blockIdx.x, b = blk >> 6, col0 = (blk & 63) << 3;
  const size_t gb = (size_t)b << 18;
  if (t < 32)
    tdm_load((uint32_t)(uintptr_t)&tile[0], ws + gb + col0, 8u, 512u, 512u, 512u, 512u);
  fill_tables(t512, tN, t);                      // overlap with DMA
  if (t < 32) wait_tensor0();
  __syncthreads();
  wave_fft512(tile + wave, 8, 1.0f, lane, t512);
  __syncthreads();
  for (int i = t; i < 4096; i += 256) {
    int row = i >> 3, c = i & 7;
    size_t n = gb + ((size_t)row << 9) + col0 + c;
    float2 q = tile[i];
    out[n]      = q.x * scale;
    out[BN + n] = q.y * scale;
  }
}

// ---------------------------------------------------------------------------
extern "C" void kernel_launch(void* const* d_in, const int* in_sizes, int n_in,
                              void* d_out, int out_size, void* d_ws,
                              size_t ws_size, hipStream_t stream) {
  const float* xr = (const float*)d_in[0];
  const float* xi = (const float*)d_in[1];
  float* out = (float*)d_out;
  float2* ws = (float2*)d_ws;   // needs BATCH*NTOT complex64 = 8 MB

  const float dz      = 80.0f / 64.0f;
  const float FOURPI2 = 39.4784176043574f;     // (2*pi)^2
  const float NLC     = -1.27f * dz;           // -gamma*dz

  kc_first<<<256, 256, 0, stream>>>(xr, xi, ws);
  for (int s = 0; s < 65; ++s) {
    const float h = (s == 0 || s == 64) ? 0.5f * dz : dz;
    const float dcoef = 10.835f * h * FOURPI2; // -(beta2/2)*h*(2*pi)^2
    kr_step<<<256, 256, 0, stream>>>(ws, dcoef);
    if (s < 64) {
      const float hprev = (s == 0) ? 0.5f * dz : dz;
      const float scale = expf(-0.023f * hprev) / 262144.0f;  // atten/N
      kc_mid<<<256, 256, 0, stream>>>(ws, scale, NLC);
    } else {
      const float scale = expf(-0.023f * 0.5f * dz) / 262144.0f;
      kc_last<<<256, 256, 0, stream>>>(ws, out, scale);
    }
  }
}